// SDE_77369540870300
// MI455X (gfx1250) — compile-verified
//
#include <hip/hip_runtime.h>
#include <cstdint>

typedef __attribute__((ext_vector_type(4))) float v4f;
typedef __attribute__((ext_vector_type(4))) int   v4i;

// Pointer types matching __builtin_amdgcn_global_load_async_to_lds_b128:
//   param0: int4 addrspace(1)* (global source)
//   param1: int4 addrspace(3)* (LDS destination)
typedef __attribute__((address_space(1))) v4i* gas_b128_t;
typedef __attribute__((address_space(3))) v4i* las_b128_t;

#ifndef __has_builtin
#define __has_builtin(x) 0
#endif

#if __has_builtin(__builtin_amdgcn_global_load_async_to_lds_b128)
#define USE_ASYNC_LDS 1
#else
#define USE_ASYNC_LDS 0
#endif

#if __has_builtin(__builtin_amdgcn_s_wait_asynccnt)
#define WAIT_ASYNCCNT(n) __builtin_amdgcn_s_wait_asynccnt(n)
#else
#define WAIT_ASYNCCNT(n) asm volatile("s_wait_asynccnt %0" ::"i"(n) : "memory")
#endif

#define TPB 256       // 8 wave32s per block
#define VEC 4         // paths per thread (b128 streaming)
#define PIPE 8        // async prefetch depth (timesteps in flight per wave)
#define MAXSTEPS 512  // coef LDS capacity (steps = 200 here)
#define CPOL_NT 1     // gfx12 CPol: TH[2:0]=1 -> non-temporal (single-use stream)

__device__ __forceinline__ void upd(v4f& y, const v4f z, const float a, const float b) {
    // y <- y + y*(a + b*z)  ==  y + mu*y*dt + sigma*y*dW   (dW = z*sqrt(dt))
    y.x = fmaf(y.x, fmaf(b, z.x, a), y.x);
    y.y = fmaf(y.y, fmaf(b, z.y, a), y.y);
    y.z = fmaf(y.z, fmaf(b, z.z, a), y.z);
    y.w = fmaf(y.w, fmaf(b, z.w, a), y.w);
}

__global__ __launch_bounds__(TPB) void sde_em_kernel(
    const float* __restrict__ y0,
    const float* __restrict__ times,
    const float* __restrict__ noise,
    const float* __restrict__ mu_p,
    const float* __restrict__ sigma_p,
    float* __restrict__ out,
    const int S, const int steps)
{
    __shared__ float2 coef[MAXSTEPS];   // (mu*dt, sigma*sqrt(dt)) per step
#if USE_ASYNC_LDS
    __shared__ v4f zbuf[PIPE][TPB];     // 8 stages x 4KB noise staging per block
#endif
    const float mu    = mu_p[0];
    const float sigma = sigma_p[0];

    const int nsteps = steps < MAXSTEPS ? steps : MAXSTEPS;
    for (int t = threadIdx.x; t < nsteps; t += TPB) {
        const float dt = times[t + 1] - times[t];
        coef[t] = make_float2(mu * dt, sigma * sqrtf(dt));
    }
    __syncthreads();

    const long long base = ((long long)blockIdx.x * TPB + threadIdx.x) * VEC;

    if (((S & (VEC - 1)) == 0) && (base + VEC <= (long long)S)) {
        const v4f yinit = *(const v4f*)(y0 + base);
        v4f y = yinit;
        const float* __restrict__ g = noise + base;   // stride S floats per step

#if USE_ASYNC_LDS
        // ---- prologue: fill the async pipeline ----
        const int npre = steps < PIPE ? steps : PIPE;
        for (int k = 0; k < npre; ++k) {
            __builtin_amdgcn_global_load_async_to_lds_b128(
                (gas_b128_t)(g + (long long)k * S),
                (las_b128_t)&zbuf[k][threadIdx.x],
                0, CPOL_NT);
        }
        int t = 0;
        const int nmain = steps > PIPE ? steps - PIPE : 0;
        // ---- steady state: wait-oldest, consume, refill same slot ----
        #pragma unroll 4
        for (; t < nmain; ++t) {
            WAIT_ASYNCCNT(PIPE - 1);                 // oldest copy (step t) landed in LDS
            const v4f z = zbuf[t & (PIPE - 1)][threadIdx.x];
            const float2 c = coef[t];
            upd(y, z, c.x, c.y);
            __builtin_amdgcn_global_load_async_to_lds_b128(
                (gas_b128_t)(g + (long long)(t + PIPE) * S),
                (las_b128_t)&zbuf[t & (PIPE - 1)][threadIdx.x],
                0, CPOL_NT);
        }
        // ---- epilogue: drain pipeline ----
        WAIT_ASYNCCNT(0);
        for (; t < steps; ++t) {
            const v4f z = zbuf[t & (PIPE - 1)][threadIdx.x];
            const float2 c = coef[t];
            upd(y, z, c.x, c.y);
        }
#else
        // Fallback: register-pipelined non-temporal streaming loads.
        v4f zc = __builtin_nontemporal_load((const v4f*)g);
        #pragma unroll 2
        for (int t = 0; t < steps - 1; ++t) {
            const v4f zn = __builtin_nontemporal_load((const v4f*)(g + (long long)(t + 1) * S));
            const float2 c = coef[t];
            upd(y, zc, c.x, c.y);
            zc = zn;
        }
        if (steps > 0) {
            const float2 c = coef[steps - 1];
            upd(y, zc, c.x, c.y);
        }
#endif
        v4f r;
        r.x = logf(y.x / yinit.x);
        r.y = logf(y.y / yinit.y);
        r.z = logf(y.z / yinit.z);
        r.w = logf(y.w / yinit.w);
        __builtin_nontemporal_store(r, (v4f*)(out + base));
    } else {
        // Scalar tail / non-multiple-of-4 safety path.
        for (long long i = base; i < (long long)S && i < base + VEC; ++i) {
            float y = y0[i];
            for (int t = 0; t < steps; ++t) {
                const float z = noise[(long long)t * S + i];
                const float2 c = (t < MAXSTEPS) ? coef[t]
                                                : make_float2(mu * (times[t + 1] - times[t]),
                                                              sigma * sqrtf(times[t + 1] - times[t]));
                y = fmaf(y, fmaf(c.y, z, c.x), y);
            }
            out[i] = logf(y / y0[i]);
        }
    }
}

extern "C" void kernel_launch(void* const* d_in, const int* in_sizes, int n_in,
                              void* d_out, int out_size, void* d_ws, size_t ws_size,
                              hipStream_t stream) {
    const float* y0    = (const float*)d_in[0];   // initial_states [S]
    const float* times = (const float*)d_in[1];   // [T]
    const float* noise = (const float*)d_in[2];   // [T-1, S]
    const float* mu    = (const float*)d_in[3];   // scalar
    const float* sigma = (const float*)d_in[4];   // scalar
    float* out = (float*)d_out;

    const int S = in_sizes[0];
    const int T = in_sizes[1];
    const int steps = T - 1;

    const int threads = (S + VEC - 1) / VEC;
    const int blocks  = (threads + TPB - 1) / TPB;
    sde_em_kernel<<<blocks, TPB, 0, stream>>>(y0, times, noise, mu, sigma, out, S, steps);
}